// DetectionPostprocess_6700148982173
// MI455X (gfx1250) — compile-verified
//
#include <hip/hip_runtime.h>
#include <hip/hip_bf16.h>
#include <stdint.h>

#define DD        24
#define AA        (DD * DD * DD)      // 13824 anchors
#define NTH       256                 // 8 wave32s
#define KPER      (AA / NTH)          // 54 elems per thread
#define KHALF     (KPER / 2)          // 27
#define TOPKN     60
#define THRESH_   0.15f
#define NMS_TH    0.05f
#define NMS_TOPKN 20
#define CAND_MAX  320
#define NBINS     256
#define STRIDE_   4.0f                // 96/24 on every axis

// CDNA5 async global->LDS DMA path (gfx1250). Proven present on this toolchain
// (round-1 diagnostic came from inside this block), but keep the guard.
#if __has_builtin(__builtin_amdgcn_global_load_async_to_lds_b32) && \
    __has_builtin(__builtin_amdgcn_s_wait_asynccnt)
#define USE_ASYNC_LDS 1
typedef __attribute__((address_space(1))) int async_gint_t;  // global
typedef __attribute__((address_space(3))) int async_lint_t;  // LDS
#endif

// Order-preserving float -> u32 (jax top_k tie-break handled by | ~idx below).
__device__ __forceinline__ uint32_t f2ord(float f) {
  uint32_t b = __float_as_uint(f);
  return (b & 0x80000000u) ? ~b : (b | 0x80000000u);
}
__device__ __forceinline__ float ord2f(uint32_t u) {
  uint32_t b = (u & 0x80000000u) ? (u ^ 0x80000000u) : ~u;
  return __uint_as_float(b);
}

__device__ __forceinline__ int bin_of(float x) {
  int bin = (int)floorf((x + 8.0f) * 16.0f);   // covers N(0,1) logits easily
  return min(max(bin, 0), NBINS - 1);
}

__global__ __launch_bounds__(NTH) void detpost_kernel(
    const float* __restrict__ cls,
    const float* __restrict__ shp,
    const float* __restrict__ off,
    float* __restrict__ out)
{
  __shared__ float              s_logit[AA];        // 55296 B
  __shared__ int                s_hist[NBINS];      //  1024 B
  __shared__ unsigned long long s_cand[CAND_MAX];   //  2560 B
  __shared__ unsigned long long s_red[NTH / 32];    //    64 B
  __shared__ unsigned long long s_best;
  __shared__ int                s_ncand;
  __shared__ int                s_bstar;
  __shared__ int                s_topidx[TOPKN];
  __shared__ float              s_toplogit[TOPKN];
  __shared__ float              s_ctr[TOPKN][3];
  __shared__ float              s_siz[TOPKN][3];
  __shared__ float              s_sc[TOPKN];
  __shared__ unsigned long long s_sup[TOPKN];
  __shared__ unsigned long long s_kept;
  __shared__ int                s_nkept;

  const int tid  = threadIdx.x;
  const int b    = blockIdx.x;
  const int wid  = tid >> 5;
  const int lane = tid & 31;

  // ---- Phase 1a: issue all async DMA loads first (global -> LDS) ----------
  const float* gin = cls + (size_t)b * AA;
#if USE_ASYNC_LDS
  #pragma unroll 6
  for (int k = 0; k < KPER; ++k) {
    int a = tid + k * NTH;
    __builtin_amdgcn_global_load_async_to_lds_b32(
        (async_gint_t*)(gin + a),
        (async_lint_t*)(&s_logit[a]),
        0, 0);
  }
#endif

  // ---- Phase 1b: small LDS inits execute under the in-flight DMA ----------
  if (tid < NBINS) s_hist[tid] = 0;
  for (int c = tid; c < CAND_MAX; c += NTH) s_cand[c] = 0ull;
  if (tid == 0) s_ncand = 0;
  if (tid < TOPKN) s_sup[tid] = 0ull;
  __syncthreads();   // hist init visible before any atomicAdd; DMA keeps flowing

  // ---- Phase 2: histogram, software-pipelined against the DMA -------------
  // Async loads complete in order per wave; each thread reads only LDS words
  // written by its own wave's DMA ops, so a partial asynccnt wait suffices.
#if USE_ASYNC_LDS
  __builtin_amdgcn_s_wait_asynccnt(KPER - KHALF);   // chunks [0,27) landed
  for (int k = 0; k < KHALF; ++k) {
    int a = tid + k * NTH;
    atomicAdd(&s_hist[bin_of(s_logit[a])], 1);
  }
  __builtin_amdgcn_s_wait_asynccnt(0);              // chunks [27,54) landed
  for (int k = KHALF; k < KPER; ++k) {
    int a = tid + k * NTH;
    atomicAdd(&s_hist[bin_of(s_logit[a])], 1);
  }
#else
  for (int k = 0; k < KPER; ++k) {
    int a = tid + k * NTH;
    float x = gin[a];
    s_logit[a] = x;
    atomicAdd(&s_hist[bin_of(x)], 1);
  }
#endif
  __syncthreads();

  if (tid == 0) {
    int cum = 0, bstar = 0;
    for (int i = NBINS - 1; i >= 0; --i) {
      cum += s_hist[i];
      if (cum >= TOPKN) { bstar = i; break; }
    }
    s_bstar = bstar;
  }
  __syncthreads();
  const int bstar = s_bstar;

  // ---- Phase 3: compact candidates (value,index) keys ---------------------
  for (int k = 0; k < KPER; ++k) {
    int a = tid + k * NTH;
    float x = s_logit[a];
    if (bin_of(x) >= bstar) {
      int slot = atomicAdd(&s_ncand, 1);
      if (slot < CAND_MAX) {
        unsigned long long key =
            ((unsigned long long)f2ord(x) << 32) |
            (unsigned long long)(0xFFFFFFFFu - (uint32_t)a);   // lower idx wins ties
        s_cand[slot] = key;
      }
    }
  }
  __syncthreads();
  const int ncand = min(s_ncand, CAND_MAX);

  // ---- Phase 4: exact top-60 via iterative block argmax (wave32 shuffles) -
  for (int sel = 0; sel < TOPKN; ++sel) {
    unsigned long long lbest = 0ull;
    int lslot = -1;
    for (int c = tid; c < ncand; c += NTH) {
      unsigned long long kk = s_cand[c];
      if (kk > lbest) { lbest = kk; lslot = c; }
    }
    unsigned long long w = lbest;
    #pragma unroll
    for (int o = 16; o > 0; o >>= 1) {
      unsigned long long t = __shfl_xor(w, o, 32);
      if (t > w) w = t;
    }
    if (lane == 0) s_red[wid] = w;
    __syncthreads();
    if (tid == 0) {
      unsigned long long bb = 0ull;
      #pragma unroll
      for (int i = 0; i < NTH / 32; ++i) if (s_red[i] > bb) bb = s_red[i];
      s_best = bb;
      int   idx = (bb == 0ull) ? 0 : (int)(0xFFFFFFFFu - (uint32_t)(bb & 0xFFFFFFFFull));
      float lg  = (bb == 0ull) ? -1e30f : ord2f((uint32_t)(bb >> 32));
      s_topidx[sel]   = idx;
      s_toplogit[sel] = lg;
    }
    __syncthreads();
    if (lslot >= 0 && lbest == s_best && s_best != 0ull) s_cand[lslot] = 0ull;
    __syncthreads();
  }

  // ---- Phase 5: gather box params for the 60 winners ----------------------
  if (tid < TOPKN) {
    int idx = s_topidx[tid];
    float lg = s_toplogit[tid];
    s_sc[tid] = 1.0f / (1.0f + __expf(-lg));
    int z = idx / (DD * DD);
    int rem = idx - z * DD * DD;
    int y = rem / DD;
    int x = rem - y * DD;
    const size_t base = (size_t)b * 3 * AA;
    float o0 = off[base + 0 * AA + idx];
    float o1 = off[base + 1 * AA + idx];
    float o2 = off[base + 2 * AA + idx];
    float h0 = shp[base + 0 * AA + idx];
    float h1 = shp[base + 1 * AA + idx];
    float h2 = shp[base + 2 * AA + idx];
    s_ctr[tid][0] = ((float)z + o0) * STRIDE_;
    s_ctr[tid][1] = ((float)y + o1) * STRIDE_;
    s_ctr[tid][2] = ((float)x + o2) * STRIDE_;
    s_siz[tid][0] = 2.0f * h0;
    s_siz[tid][1] = 2.0f * h1;
    s_siz[tid][2] = 2.0f * h2;
  }
  __syncthreads();

  // ---- Phase 6: 60x60 IoU -> per-row suppression bitmasks -----------------
  for (int p = tid; p < TOPKN * TOPKN; p += NTH) {
    int i = p / TOPKN, j = p - i * TOPKN;
    if (i == j) continue;
    float inter = 1.0f, voli = 1.0f, volj = 1.0f;
    #pragma unroll
    for (int k = 0; k < 3; ++k) {
      float ci = s_ctr[i][k], cj = s_ctr[j][k];
      float si = s_siz[i][k], sj = s_siz[j][k];
      float ov = fminf(ci + 0.5f * si, cj + 0.5f * sj) -
                 fmaxf(ci - 0.5f * si, cj - 0.5f * sj);
      inter *= fmaxf(ov, 0.0f);
      voli  *= si;
      volj  *= sj;
    }
    float uni = voli + volj - inter;
    float iou = inter / fmaxf(uni, 1e-8f);
    if (iou > NMS_TH) atomicOr(&s_sup[i], 1ull << j);   // ds_or_b64
  }
  __syncthreads();

  // ---- Phase 7: greedy NMS on bitmasks (serial, 60 steps) -----------------
  if (tid == 0) {
    unsigned long long kept = 0ull, supp = 0ull;
    int nvalid = 0;
    for (int i = 0; i < TOPKN; ++i) {
      bool valid = s_sc[i] > THRESH_;
      bool cand  = valid && (nvalid < NMS_TOPKN);   // rank = nvalid before i
      if (valid) ++nvalid;
      bool keep = cand && !((supp >> i) & 1ull);
      if (keep) {
        kept |= 1ull << i;
        supp |= s_sup[i];
      }
    }
    s_kept  = kept;
    s_nkept = __popcll(kept);
  }
  __syncthreads();

  // ---- Phase 8: emit rows: kept (score order) first, then -1 fill ---------
  if (tid < TOPKN) {
    float* row = out + ((size_t)b * TOPKN + tid) * 8;
    if (tid < s_nkept) {
      unsigned long long kept = s_kept;
      int i = 0, cnt = -1;
      for (int q = 0; q < TOPKN; ++q) {
        if ((kept >> q) & 1ull) {
          if (++cnt == tid) { i = q; break; }
        }
      }
      row[0] = 1.0f;
      row[1] = s_sc[i];
      row[2] = s_ctr[i][0]; row[3] = s_ctr[i][1]; row[4] = s_ctr[i][2];
      row[5] = s_siz[i][0]; row[6] = s_siz[i][1]; row[7] = s_siz[i][2];
    } else {
      #pragma unroll
      for (int c = 0; c < 8; ++c) row[c] = -1.0f;
    }
  }
}

extern "C" void kernel_launch(void* const* d_in, const int* in_sizes, int n_in,
                              void* d_out, int out_size, void* d_ws, size_t ws_size,
                              hipStream_t stream) {
  (void)n_in; (void)out_size; (void)d_ws; (void)ws_size;
  const float* cls = (const float*)d_in[0];   // (B,1,24,24,24)
  const float* shp = (const float*)d_in[1];   // (B,3,24,24,24)
  const float* off = (const float*)d_in[2];   // (B,3,24,24,24)
  float* out = (float*)d_out;                 // (B,60,8)
  const int B = in_sizes[0] / AA;
  detpost_kernel<<<dim3(B), dim3(NTH), 0, stream>>>(cls, shp, off, out);
}